// SupAux_30545807409307
// MI455X (gfx1250) — compile-verified
//
#include <hip/hip_runtime.h>
#include <stdint.h>

#define NB 4
#define NC 64
#define HW 262144            // 512*512
#define NSEG 1024
#define BKSEG (NB * NSEG)    // 4096 segments total
#define WEIGHT_F 0.1f
#define EPS_F 1e-5f
#define SLICES 64
#define SLICE_PIX (HW / SLICES)   // 4096 pixels per block
#define TPB 1024

typedef __attribute__((ext_vector_type(4))) unsigned int v4u_t;
typedef __attribute__((ext_vector_type(8))) int          v8i_t;
typedef __attribute__((ext_vector_type(4))) int          v4i_t;

// ---------------------------------------------------------------- init
__global__ void k_init(float* ws, int n) {
    int i = blockIdx.x * blockDim.x + threadIdx.x;
    if (i < n) ws[i] = 0.0f;   // zero bit pattern also zeroes the int maxlabel slot
}

// ---------------------------------------------------------------- accumulate
// One block = one (batch, pixel-slice). Whole per-batch accumulator lives in LDS:
//   lsum[64][1024] (256KB, channel-major so ds_add bank = label%64, conflict-free-ish)
//   lcnt[1024]     (4KB)
//   llab[4096]     (16KB)  <- filled by the Tensor Data Mover
// Total 276KB dynamic LDS (fits CDNA5's 320KB WGP LDS).
__global__ void __launch_bounds__(TPB)
k_accum(const float* __restrict__ inp, const int* __restrict__ sp,
        float* __restrict__ sums, float* __restrict__ counts,
        int* __restrict__ maxlab) {
    extern __shared__ char smem[];
    float* lsum = (float*)smem;                                   // [NC][NSEG]
    float* lcnt = (float*)(smem + (size_t)NC * NSEG * 4);         // [NSEG]
    int*   llab = (int*)(smem + (size_t)NC * NSEG * 4 + NSEG * 4);// [SLICE_PIX]

    const int t    = threadIdx.x;
    const int b    = blockIdx.x / SLICES;
    const int base = (blockIdx.x % SLICES) * SLICE_PIX;

#if defined(__gfx1250__) && __has_builtin(__builtin_amdgcn_tensor_load_to_lds)
    // --- TDM: DMA the 16KB label tile global->LDS (wave 0 issues; EXEC-independent)
    if (t < 32) {
        unsigned long long ga = (unsigned long long)(uintptr_t)(sp + (size_t)b * HW + base);
        unsigned ldsoff = (unsigned)(uintptr_t)llab;   // low 32 bits of generic LDS ptr = LDS byte offset
        // D# group0: count=1 | lds_addr | global_addr[56:0] | type=2
        v4u_t g0 = { 1u,
                     ldsoff,
                     (unsigned)(ga & 0xFFFFFFFFull),
                     (unsigned)((ga >> 32) & 0x01FFFFFFull) | (2u << 30) };
        // D# group1: data_size=4B; tensor_dim0=4096, tensor_dim1=1;
        //            tile_dim0=4096, tile_dim1=1; tensor_dim0_stride=4096
        v8i_t g1 = { (int)(2u << 16),                         // mask=0, data_size=2 (4B)
                     (int)((unsigned)(SLICE_PIX & 0xFFFF) << 16),   // tensor_dim0 lo16 -> [63:48]
                     (int)(((unsigned)SLICE_PIX >> 16) | (1u << 16)),// tensor_dim0 hi16, tensor_dim1=1 lo16
                     (int)((unsigned)(SLICE_PIX & 0xFFFF) << 16),   // tile_dim0 -> [127:112]
                     1,                                             // tile_dim1=1, tile_dim2=0
                     SLICE_PIX,                                     // tensor_dim0_stride lo32
                     0, 0 };
        v4i_t gz4  = { 0, 0, 0, 0 };
        v8i_t gz8  = { 0, 0, 0, 0, 0, 0, 0, 0 };
        __builtin_amdgcn_tensor_load_to_lds(g0, g1, gz4, gz4, gz8, 0);
    }
#else
    for (int i = t; i < SLICE_PIX; i += TPB) llab[i] = sp[(size_t)b * HW + base + i];
#endif

    // zero accumulators (disjoint LDS range; overlaps with the TDM transfer)
    for (int i = t; i < NC * NSEG; i += TPB) lsum[i] = 0.0f;
    for (int i = t; i < NSEG; i += TPB)      lcnt[i] = 0.0f;

#if defined(__gfx1250__) && __has_builtin(__builtin_amdgcn_tensor_load_to_lds)
    if (t < 32) __builtin_amdgcn_s_wait_tensorcnt(0);
#endif
    __syncthreads();

    // each thread owns 4 consecutive pixels: coalesced float4 feature loads
    const int4 labs = ((const int4*)llab)[t];
    int lmax = max(max(labs.x, labs.y), max(labs.z, labs.w));
    atomicMax(maxlab, lmax);

    const float4* inp4 = (const float4*)inp;
    const int rowBase = (b * NC) * (HW / 4) + (base / 4) + t;
    #pragma unroll 4
    for (int c = 0; c < NC; ++c) {
        float4 v = inp4[rowBase + c * (HW / 4)];
        atomicAdd(&lsum[c * NSEG + labs.x], v.x);   // ds_add_f32, bank = label%64
        atomicAdd(&lsum[c * NSEG + labs.y], v.y);
        atomicAdd(&lsum[c * NSEG + labs.z], v.z);
        atomicAdd(&lsum[c * NSEG + labs.w], v.w);
    }
    atomicAdd(&lcnt[labs.x], 1.0f);
    atomicAdd(&lcnt[labs.y], 1.0f);
    atomicAdd(&lcnt[labs.z], 1.0f);
    atomicAdd(&lcnt[labs.w], 1.0f);
    __syncthreads();

    // flush: thread t owns label t (TPB == NSEG); transpose LDS [c][lab] -> global [seg][c]
    {
        const int lab = t;
        float cv = lcnt[lab];
        if (cv != 0.0f) unsafeAtomicAdd(&counts[b * NSEG + lab], cv);
        float* dst = &sums[(size_t)(b * NSEG + lab) * NC];
        #pragma unroll 8
        for (int c = 0; c < NC; ++c) {
            float v = lsum[c * NSEG + lab];
            if (v != 0.0f) unsafeAtomicAdd(&dst[c], v);
        }
    }
}

// ---------------------------------------------------------------- finalize mean
__global__ void k_mean(float* __restrict__ sums, const float* __restrict__ counts) {
    int i = blockIdx.x * blockDim.x + threadIdx.x;
    if (i < BKSEG * NC) sums[i] = sums[i] / (counts[i / NC] + EPS_F);
}

// ---------------------------------------------------------------- apply update
__global__ void k_apply(const float* __restrict__ inp, const int* __restrict__ sp,
                        const float* __restrict__ mean, const int* __restrict__ maxlab,
                        float* __restrict__ out) {
    const int total4 = NB * NC * HW / 4;
    int i4 = blockIdx.x * blockDim.x + threadIdx.x;
    if (i4 >= total4) return;
    int e = i4 * 4;
    int p = e & (HW - 1);          // HW = 2^18
    int c = (e >> 18) & (NC - 1);
    int b = e >> 24;               // e / (NC*HW)
    float4 v  = ((const float4*)inp)[i4];
    int4 lab  = ((const int4*)sp)[((size_t)b * HW + p) >> 2];
    int ml    = *maxlab;           // strict '<': pixel holding the max label gets no update
    const float* mb = mean + (size_t)b * NSEG * NC + c;
    if (lab.x < ml) v.x += WEIGHT_F * mb[(size_t)lab.x * NC];
    if (lab.y < ml) v.y += WEIGHT_F * mb[(size_t)lab.y * NC];
    if (lab.z < ml) v.z += WEIGHT_F * mb[(size_t)lab.z * NC];
    if (lab.w < ml) v.w += WEIGHT_F * mb[(size_t)lab.w * NC];
    ((float4*)out)[i4] = v;
}

// ---------------------------------------------------------------- launcher
extern "C" void kernel_launch(void* const* d_in, const int* in_sizes, int n_in,
                              void* d_out, int out_size, void* d_ws, size_t ws_size,
                              hipStream_t stream) {
    const float* inp = (const float*)d_in[0];
    const int*   sp  = (const int*)d_in[1];
    float*       out = (float*)d_out;

    // workspace layout: sums [4096*64] | counts [4096] | maxlabel [1]
    float* sums   = (float*)d_ws;
    float* counts = sums + BKSEG * NC;
    int*   maxlab = (int*)(counts + BKSEG);
    const int initN = BKSEG * NC + BKSEG + 1;
    if (ws_size < (size_t)initN * 4) return;

    k_init<<<(initN + 255) / 256, 256, 0, stream>>>((float*)d_ws, initN);

    const size_t shbytes = (size_t)NC * NSEG * 4 + NSEG * 4 + SLICE_PIX * 4; // 276KB
    k_accum<<<NB * SLICES, TPB, shbytes, stream>>>(inp, sp, sums, counts, maxlab);

    k_mean<<<(BKSEG * NC + 255) / 256, 256, 0, stream>>>(sums, counts);

    const int total4 = NB * NC * HW / 4;
    k_apply<<<(total4 + 255) / 256, 256, 0, stream>>>(inp, sp, sums, maxlab, out);
}